// HeightCompression_25984552140992
// MI455X (gfx1250) — compile-verified
//
#include <hip/hip_runtime.h>

typedef float    v2f __attribute__((ext_vector_type(2)));
typedef float    v8f __attribute__((ext_vector_type(8)));
typedef unsigned v4u __attribute__((ext_vector_type(4)));
typedef int      v4i __attribute__((ext_vector_type(4)));
typedef int      v8i __attribute__((ext_vector_type(8)));

#define BLOCK   256
#define CHUNK   1024           // known points per TDM chunk (divides 2048/4096/8192)
#define BB      4
#define N_PTS   8192
#define M2      8192
#define M3      4096
#define M4      2048
#define MMATCH  4096
#define IDX_MASK 0x1FFFu       // 13 bits: max M = 8192
#define KEY_MASK 0xFFFFE000u

// ---------------------------------------------------------------------------
// TDM: issue a 1-D async copy of CHUNK*3 f32 from global to LDS (leader wave).
// D# layout per CDNA5 ISA ch.8: group0 = {flags, lds_addr, global_addr, type},
// group1 = {data_size, tensor/tile dims, stride}. Tracked by TENSORcnt.
// 6-arg builtin form (clang-23 / therock-10.0 headers).
// ---------------------------------------------------------------------------
__device__ __forceinline__ void tdm_load_chunk(const float* __restrict__ src,
                                               const float* lds_dst)
{
    const unsigned elems = CHUNK * 3;                    // 3072 f32 = 12KB
    unsigned long long ga = (unsigned long long)(uintptr_t)src;
    unsigned ldsoff = (unsigned)(uintptr_t)(const void*)lds_dst; // low 32b = LDS offset

    v4u g0;
    g0[0] = 1u;                                          // count=1, user D#, no gather
    g0[1] = ldsoff;                                      // lds_addr (bytes)
    g0[2] = (unsigned)(ga & 0xFFFFFFFFu);                // global_addr[31:0]
    g0[3] = (unsigned)((ga >> 32) & 0x01FFFFFFu) | (2u << 30); // addr[56:32] | type=2

    v8i g1;
    g1[0] = (int)(2u << 16);                             // data_size=2 (4 bytes)
    g1[1] = (int)(elems << 16);                          // tensor_dim0[15:0] @ bits63:48
    g1[2] = (int)(1u << 16);                             // tensor_dim0[31:16]=0, tensor_dim1=1
    g1[3] = (int)(elems << 16);                          // tile_dim0 @ bits127:112
    g1[4] = 1;                                           // tile_dim1=1, tile_dim2=0
    g1[5] = (int)elems;                                  // tensor_dim0_stride[31:0]
    g1[6] = 0;                                           // stride hi, dim1_stride lo
    g1[7] = 0;

    v4i g2 = {0, 0, 0, 0};
    v4i g3 = {0, 0, 0, 0};
    v8i g4 = {0, 0, 0, 0, 0, 0, 0, 0};
    __builtin_amdgcn_tensor_load_to_lds(g0, g1, g2, g3, g4, 0);
}

// branchless sorted insert into ascending (t0 <= t1 <= t2)
__device__ __forceinline__ void insert3(unsigned key, unsigned& t0, unsigned& t1, unsigned& t2) {
    unsigned m0 = (t0 > key) ? t0 : key;
    t0          = (t0 > key) ? key : t0;
    unsigned m1 = (t1 > m0)  ? t1 : m0;
    t1          = (t1 > m0)  ? m0 : t1;
    t2          = (t2 > m1)  ? m1 : t2;
}

// ---------------------------------------------------------------------------
// Scan one known set with V_WMMA_F32_16X16X4_F32, TDM double-buffered.
// A rows (known m): [-2x, -2y, -2z, |k|^2]; B cols (query q): [x, y, z, 1].
// D = |k|^2 - 2 q.k  (adding |q|^2 later is order-invariant per lane).
// ---------------------------------------------------------------------------
__device__ __forceinline__ void scan_topk(const float* __restrict__ kn, int M,
        float* s_a, float* s_b, bool leader, int tid, v2f bop, float qn,
        bool hi, int half8, unsigned& t0, unsigned& t1, unsigned& t2)
{
    const int nch = M / CHUNK;
    __syncthreads();                                   // prior buffer users done
    if (leader) tdm_load_chunk(kn, s_a);               // chunk 0 -> A
    for (int c = 0; c < nch; ++c) {
        if (leader) __builtin_amdgcn_s_wait_tensorcnt(0);
        __syncthreads();                               // chunk c visible to block
        if (leader && (c + 1 < nch))
            tdm_load_chunk(kn + (size_t)(c + 1) * CHUNK * 3, (c & 1) ? s_a : s_b);
        const float* sp = (c & 1) ? s_b : s_a;
#pragma unroll 4
        for (int t = 0; t < CHUNK; t += 16) {
            int ml = t + (tid & 15);
            float x = sp[ml * 3 + 0];
            float y = sp[ml * 3 + 1];
            float z = sp[ml * 3 + 2];
            v2f av;                                    // A 16x4 f32: lanes<16 K0,K1; >=16 K2,K3
            av.x = hi ? (-2.0f * z) : (-2.0f * x);
            av.y = hi ? (x * x + y * y + z * z) : (-2.0f * y);
            v8f cz = {};
            v8f d = __builtin_amdgcn_wmma_f32_16x16x4_f32(
                        false, av, false, bop, (short)0, cz, false, false);
            unsigned mbase = (unsigned)(c * CHUNK + t + half8);
#pragma unroll
            for (int r = 0; r < 8; ++r) {
                float v = fmaxf(d[r] + qn, 0.0f);      // monotone key: clamped squared dist
                unsigned key = (__float_as_uint(v) & KEY_MASK) | (mbase + (unsigned)r);
                insert3(key, t0, t1, t2);
            }
        }
    }
}

__device__ __forceinline__ float scan_min(const float* __restrict__ kn, int M,
        float* s_a, float* s_b, bool leader, int tid, v2f bop, bool hi)
{
    float mn = 3.0e38f;
    const int nch = M / CHUNK;
    __syncthreads();
    if (leader) tdm_load_chunk(kn, s_a);
    for (int c = 0; c < nch; ++c) {
        if (leader) __builtin_amdgcn_s_wait_tensorcnt(0);
        __syncthreads();
        if (leader && (c + 1 < nch))
            tdm_load_chunk(kn + (size_t)(c + 1) * CHUNK * 3, (c & 1) ? s_a : s_b);
        const float* sp = (c & 1) ? s_b : s_a;
#pragma unroll 4
        for (int t = 0; t < CHUNK; t += 16) {
            int ml = t + (tid & 15);
            float x = sp[ml * 3 + 0];
            float y = sp[ml * 3 + 1];
            float z = sp[ml * 3 + 2];
            v2f av;
            av.x = hi ? (-2.0f * z) : (-2.0f * x);
            av.y = hi ? (x * x + y * y + z * z) : (-2.0f * y);
            v8f cz = {};
            v8f d = __builtin_amdgcn_wmma_f32_16x16x4_f32(
                        false, av, false, bop, (short)0, cz, false, false);
#pragma unroll
            for (int r = 0; r < 8; ++r) mn = fminf(mn, d[r]);
        }
    }
    return mn;
}

// Exact re-evaluation + inverse-distance weighted projection for one level.
__device__ __forceinline__ float level_contrib(unsigned t0, unsigned t1, unsigned t2,
        const float* __restrict__ knB, const float* __restrict__ featB,
        const float* s_vseg, float qx, float qy, float qz)
{
    unsigned keys[3] = {t0, t1, t2};
    float rsum = 0.0f, acc = 0.0f;
#pragma unroll
    for (int k = 0; k < 3; ++k) {
        int idx = (int)(keys[k] & IDX_MASK);
        const float* kp = knB + (size_t)idx * 3;
        float dx = qx - kp[0], dy = qy - kp[1], dz = qz - kp[2];
        float d2 = fmaxf(dx * dx + dy * dy + dz * dz, 0.0f);
        float rc = 1.0f / (d2 + 1e-8f);
        const float* fr = featB + (size_t)idx * 32;
        float dp = 0.0f;
#pragma unroll
        for (int cc = 0; cc < 32; ++cc) dp = fmaf(fr[cc], s_vseg[cc], dp);
        rsum += rc;
        acc = fmaf(rc, dp, acc);
    }
    return acc / rsum;
}

__global__ __launch_bounds__(BLOCK, 4) void heightmap_kernel(
    const float* __restrict__ pts,   // (B,N,3)
    const float* __restrict__ kn2, const float* __restrict__ f2,
    const float* __restrict__ kn3, const float* __restrict__ f3,
    const float* __restrict__ kn4, const float* __restrict__ f4,
    const float* __restrict__ mp,    // (B,Mm,3)
    const float* __restrict__ w_fc,  // (64,96)
    const float* __restrict__ w_cls, // (1,64)
    float* __restrict__ out)         // [pred (B*N) | gt (B*N)]
{
    __shared__ float s_v[96];
    __shared__ float s_bufA[CHUNK * 3];
    __shared__ float s_bufB[CHUNK * 3];

    const int tid = threadIdx.x;
    const bool leader = (tid >> 5) == 0;             // wave 0 drives the TDM
    const int blocksPerBatch = N_PTS / 128;          // 8 waves x 16 queries
    const int b = blockIdx.x / blocksPerBatch;
    const int qtile = (blockIdx.x % blocksPerBatch) * 128 + (tid >> 5) * 16;

    // v = w_cls @ w_fc  (collapses FC 96->64->1 into one 96-vector)
    if (tid < 96) {
        float a = 0.0f;
        for (int o = 0; o < 64; ++o) a = fmaf(w_cls[o], w_fc[o * 96 + tid], a);
        s_v[tid] = a;
    }

    const int q = qtile + (tid & 15);                // both lane halves share queries
    const float* qp = pts + ((size_t)b * N_PTS + q) * 3;
    const float qx = qp[0], qy = qp[1], qz = qp[2];
    const float qn = qx * qx + qy * qy + qz * qz;
    const bool hi = (tid & 16) != 0;
    const int half8 = hi ? 8 : 0;
    v2f bop;                                         // B 4x16 f32: lanes<16 K0,K1; >=16 K2,K3
    bop.x = hi ? qz : qx;
    bop.y = hi ? 1.0f : qy;

    const float* kn2B = kn2 + (size_t)b * M2 * 3;
    const float* f2B  = f2  + (size_t)b * M2 * 32;
    const float* kn3B = kn3 + (size_t)b * M3 * 3;
    const float* f3B  = f3  + (size_t)b * M3 * 32;
    const float* kn4B = kn4 + (size_t)b * M4 * 3;
    const float* f4B  = f4  + (size_t)b * M4 * 32;
    const float* mpB  = mp  + (size_t)b * MMATCH * 3;

    float pred = 0.0f;
    unsigned t0, t1, t2, s0, s1, s2;

    // ---- level 2 ----
    t0 = t1 = t2 = 0xFFFFFFFFu;
    scan_topk(kn2B, M2, s_bufA, s_bufB, leader, tid, bop, qn, hi, half8, t0, t1, t2);
    s0 = (unsigned)__shfl_xor((int)t0, 16, 32);
    s1 = (unsigned)__shfl_xor((int)t1, 16, 32);
    s2 = (unsigned)__shfl_xor((int)t2, 16, 32);
    insert3(s0, t0, t1, t2); insert3(s1, t0, t1, t2); insert3(s2, t0, t1, t2);
    pred += level_contrib(t0, t1, t2, kn2B, f2B, s_v + 0, qx, qy, qz);

    // ---- level 3 ----
    t0 = t1 = t2 = 0xFFFFFFFFu;
    scan_topk(kn3B, M3, s_bufA, s_bufB, leader, tid, bop, qn, hi, half8, t0, t1, t2);
    s0 = (unsigned)__shfl_xor((int)t0, 16, 32);
    s1 = (unsigned)__shfl_xor((int)t1, 16, 32);
    s2 = (unsigned)__shfl_xor((int)t2, 16, 32);
    insert3(s0, t0, t1, t2); insert3(s1, t0, t1, t2); insert3(s2, t0, t1, t2);
    pred += level_contrib(t0, t1, t2, kn3B, f3B, s_v + 32, qx, qy, qz);

    // ---- level 4 ----
    t0 = t1 = t2 = 0xFFFFFFFFu;
    scan_topk(kn4B, M4, s_bufA, s_bufB, leader, tid, bop, qn, hi, half8, t0, t1, t2);
    s0 = (unsigned)__shfl_xor((int)t0, 16, 32);
    s1 = (unsigned)__shfl_xor((int)t1, 16, 32);
    s2 = (unsigned)__shfl_xor((int)t2, 16, 32);
    insert3(s0, t0, t1, t2); insert3(s1, t0, t1, t2); insert3(s2, t0, t1, t2);
    pred += level_contrib(t0, t1, t2, kn4B, f4B, s_v + 64, qx, qy, qz);

    // ---- ground-truth mask: min distance to match points ----
    float mn = scan_min(mpB, MMATCH, s_bufA, s_bufB, leader, tid, bop, hi);
    mn = fminf(mn, __shfl_xor(mn, 16, 32));
    float dmin = sqrtf(fmaxf(mn + qn, 0.0f));
    float gt = (dmin < 0.5f) ? 1.0f : 0.0f;

    if (!hi) {
        size_t g = (size_t)b * N_PTS + q;
        out[g] = pred;
        out[(size_t)BB * N_PTS + g] = gt;
    }
}

extern "C" void kernel_launch(void* const* d_in, const int* in_sizes, int n_in,
                              void* d_out, int out_size, void* d_ws, size_t ws_size,
                              hipStream_t stream) {
    (void)in_sizes; (void)n_in; (void)d_ws; (void)ws_size; (void)out_size;
    const float* pts   = (const float*)d_in[0];
    const float* kn2   = (const float*)d_in[1];
    const float* f2    = (const float*)d_in[2];
    const float* kn3   = (const float*)d_in[3];
    const float* f3    = (const float*)d_in[4];
    const float* kn4   = (const float*)d_in[5];
    const float* f4    = (const float*)d_in[6];
    const float* mp    = (const float*)d_in[7];
    const float* w_fc  = (const float*)d_in[8];
    const float* w_cls = (const float*)d_in[9];
    float* out = (float*)d_out;

    dim3 grid(BB * (N_PTS / 128));
    dim3 block(BLOCK);
    heightmap_kernel<<<grid, block, 0, stream>>>(pts, kn2, f2, kn3, f3, kn4, f4,
                                                 mp, w_fc, w_cls, out);
}